// ConvPosDivMultiHeadAttn_1786706395832
// MI455X (gfx1250) — compile-verified
//
#include <hip/hip_runtime.h>
#include <hip/hip_bf16.h>

// fp32 WMMA path on gfx1250: V_WMMA_F32_16X16X4_F32
// A/B operands: 16x4 / 4x16 f32 = 2 VGPRs per lane; C/D: 16x16 f32 = 8 VGPRs.
typedef __attribute__((ext_vector_type(2))) float v2f;
typedef __attribute__((ext_vector_type(8))) float v8f;
typedef __attribute__((ext_vector_type(4))) unsigned int u32x4;
typedef __attribute__((ext_vector_type(8))) int i32x8;
typedef __attribute__((ext_vector_type(4))) int i32x4;

__device__ __forceinline__ v8f wmma4(v2f a, v2f b, v8f c) {
    // 8 args: (neg_a, A, neg_b, B, c_mod, C, reuse_a, reuse_b)
    return __builtin_amdgcn_wmma_f32_16x16x4_f32(false, a, false, b, (short)0, c,
                                                 false, false);
}

__device__ __forceinline__ v2f ld2(const float* p) {
    return *reinterpret_cast<const v2f*>(p);
}

// Tensor Data Mover: 2D fp32 tile (tile_rows x tile_cols) from a row-major
// (tensor_h x tensor_w) matrix into LDS (packed, row-major, no padding).
// D# layout per CDNA5 ISA 8.3/8.4 (group0: count/lds/global/type=2,
// group1: data_size=4B, dims, tile, dim0 stride). Groups 2/3 zero (2D tile).
// This toolchain exposes the 6-arg builtin:
//   (uint32x4 g0, int32x8 g1, int32x4 g2, int32x4 g3, int32x8 extra, i32 cpol)
__device__ __forceinline__ void tdm_load_tile(unsigned lds_addr, const void* gptr,
                                              unsigned tile_cols, unsigned tile_rows,
                                              unsigned tensor_w, unsigned tensor_h,
                                              unsigned row_stride_elems) {
    unsigned long long ga = (unsigned long long)(size_t)gptr;
    u32x4 g0;
    g0[0] = 1u;                                   // count=1, user descriptor
    g0[1] = lds_addr;                             // LDS byte address
    g0[2] = (unsigned)ga;                         // global addr [31:0]
    g0[3] = (unsigned)((ga >> 32) & 0x01ffffffu)  // global addr [56:32]
            | 0x80000000u;                        // type=2 ("image")
    i32x8 g1;
    g1[0] = 0x00020000;                           // data_size=2 -> 4 bytes
    g1[1] = (int)((tensor_w & 0xffffu) << 16);                        // dim0 lo16
    g1[2] = (int)(((tensor_w >> 16) & 0xffffu) |
                  ((tensor_h & 0xffffu) << 16));                      // dim0 hi / dim1 lo
    g1[3] = (int)(((tensor_h >> 16) & 0xffffu) |
                  ((tile_cols & 0xffffu) << 16));                     // dim1 hi / tile_dim0
    g1[4] = (int)(tile_rows & 0xffffu);                               // tile_dim1 (tile_dim2=0)
    g1[5] = (int)row_stride_elems;                                    // dim0_stride lo32
    g1[6] = 0;                                                        // stride hi / dim1_stride
    g1[7] = 0;
    i32x4 gz4 = {0, 0, 0, 0};
    i32x8 gz8 = {0, 0, 0, 0, 0, 0, 0, 0};
    __builtin_amdgcn_tensor_load_to_lds(g0, g1, gz4, gz4, gz8, 0);
}

// ---------------------------------------------------------------------------
// Sizes
#define BB 8
#define LL 512
#define DD 1024
#define HH 16
#define HD 64
#define D3 3072
#define D2 2048

// ---------------------------------------------------------------------------
// Kernel 1: qkv = x @ w_qkv, scattered to q/k/v in (B,H,L,HD) layout.
// Block = 256 threads = 8 waves computing a 32(M) x 64(N) macro-tile; the
// shared 32(K) x 64(N) B-slab is TDM-staged into LDS, double-buffered.
__global__ void k_qkv_proj(const float* __restrict__ x,
                           const float* __restrict__ w_qkv,
                           float* __restrict__ qo, float* __restrict__ ko,
                           float* __restrict__ vo) {
    __shared__ float Bs[2][32 * 64];
    int tid = threadIdx.x;
    int wave = tid >> 5, lane = tid & 31;
    int mB = blockIdx.x / 48, nB = blockIdx.x % 48;   // 128 x 48 blocks
    int m0 = mB << 5, n0 = nB << 6;
    int wm = wave >> 2, wn = wave & 3;                // 2 x 4 wave grid
    int ml = lane & 15, koff = (lane >> 4) << 1, mo = (lane >> 4) << 3;

    unsigned lds0 = (unsigned)(size_t)&Bs[0][0];
    unsigned lds1 = (unsigned)(size_t)&Bs[1][0];

    if (wave == 0)
        tdm_load_tile(lds0, w_qkv + n0, 64, 32, D3, DD, D3);

    const float* arow = x + (size_t)(m0 + (wm << 4) + ml) * DD + koff;
    v8f c = {};
    for (int kc = 0; kc < 32; ++kc) {
        if (wave == 0) {
            if (kc < 31) {
                tdm_load_tile((kc & 1) ? lds0 : lds1,
                              w_qkv + (size_t)(kc + 1) * 32 * D3 + n0,
                              64, 32, D3, DD, D3);
                __builtin_amdgcn_s_wait_tensorcnt(1);   // chunk kc landed
            } else {
                __builtin_amdgcn_s_wait_tensorcnt(0);
            }
        }
        __syncthreads();
        const float* bs = &Bs[kc & 1][0] + (wn << 4) + ml;
        int kbase = kc << 5;
        for (int kk = 0; kk < 32; kk += 4) {
            v2f a = ld2(arow + kbase + kk);
            v2f bv;
            bv.x = bs[(kk + koff) << 6];
            bv.y = bs[(kk + koff + 1) << 6];
            c = wmma4(a, bv, c);
        }
        __syncthreads();
    }

    int ncol = n0 + (wn << 4);
    int which = ncol >> 10;                // 0=q 1=k 2=v
    int d0 = ncol & 1023;
    int h = d0 >> 6, hd0 = d0 & 63;
    float* outb = (which == 0) ? qo : ((which == 1) ? ko : vo);
    int mrow0 = m0 + (wm << 4);
    for (int r = 0; r < 8; ++r) {
        int m = mrow0 + r + mo;
        int b = m >> 9, l = m & 511;
        outb[((((size_t)b * HH + h) * LL + l) * HD) + hd0 + ml] = c[r];
    }
}

// ---------------------------------------------------------------------------
// Kernel 2: sinusoid PE (computed on the fly) @ w_qkpos -> q_p/k_p (H,L,HD).
__global__ void k_pos(const float* __restrict__ w_qkpos,
                      float* __restrict__ qp, float* __restrict__ kp) {
    int id = blockIdx.x * blockDim.x + threadIdx.x;   // < L * 2D
    int l = id >> 11;
    int cc = id & 2047;
    float r = (float)l - 256.0f;
    const float C = -0.29710775379146967f;            // -ln(10000)/(32-1)
    float sum = 0.0f;
    for (int kk = 0; kk < HD; ++kk) {
        int j = kk & 31;
        float inv = __expf((float)j * C);
        float ang = r * inv;
        float pe = (kk < 32) ? __sinf(ang) : __cosf(ang);
        sum += pe * w_qkpos[kk * D2 + cc];
    }
    if (cc < DD) {
        qp[(((cc >> 6) * LL + l) * HD) + (cc & 63)] = sum;
    } else {
        int c2 = cc - DD;
        kp[(((c2 >> 6) * LL + l) * HD) + (c2 & 63)] = sum;
    }
}

// ---------------------------------------------------------------------------
// Kernel 3: fused attention for one (b,h,16-row block).
// scores = QK^T + QpKp^T (WMMA), speaker-zero, key-mask, Gaussian bias,
// softmax in LDS, then PV (WMMA) -> ctx (B,L,D).
__global__ void k_attn(const float* __restrict__ q, const float* __restrict__ k,
                       const float* __restrict__ v,
                       const float* __restrict__ qp, const float* __restrict__ kp,
                       const int* __restrict__ mask, const int* __restrict__ qmask,
                       const float* __restrict__ shift, const float* __restrict__ bias,
                       float* __restrict__ ctx) {
    __shared__ float sc[16 * 520];
    __shared__ float rmax[16], rinv[16];

    int ib = blockIdx.x & 31;              // 32 row-blocks of 16
    int bh = blockIdx.x >> 5;              // b*16 + h
    int h = bh & 15, b = bh >> 4;
    int i0 = ib << 4;
    int tid = threadIdx.x;
    int wave = tid >> 5, lane = tid & 31;
    int ml   = lane & 15;
    int koff = (lane >> 4) << 1;
    int mo   = (lane >> 4) << 3;

    const float* qb  = q  + (size_t)bh * LL * HD;
    const float* kb  = k  + (size_t)bh * LL * HD;
    const float* vb  = v  + (size_t)bh * LL * HD;
    const float* qpb = qp + (size_t)h * LL * HD;
    const float* kpb = kp + (size_t)h * LL * HD;
    float shift0 = shift[0], bias0 = bias[0];

    // --- scores: each wave does 4 j-tiles ---
    const float* qrow  = qb  + (i0 + ml) * HD + koff;
    const float* qprow = qpb + (i0 + ml) * HD + koff;
    for (int t = 0; t < 4; ++t) {
        int j0 = (wave + (t << 3)) << 4;
        const float* krow  = kb  + (j0 + ml) * HD + koff;
        const float* kprow = kpb + (j0 + ml) * HD + koff;
        v8f c = {};
        for (int kk = 0; kk < HD; kk += 4) {
            c = wmma4(ld2(qrow + kk),  ld2(krow + kk),  c);   // content
            c = wmma4(ld2(qprow + kk), ld2(kprow + kk), c);   // positional
        }
        int j = j0 + ml;
        int km  = mask[b * LL + j];
        int qmj = qmask[b * LL + j];
        for (int r = 0; r < 8; ++r) {
            int mrow = r + mo;
            int irow = i0 + mrow;
            float val = c[r];
            int qmi = qmask[b * LL + irow];
            val = (qmi == qmj) ? val : 0.0f;   // speaker-identity zeroing
            val = (km == 0) ? -1e30f : val;    // key padding mask
            float dij = (float)(irow - j);
            val -= shift0 * dij * dij + bias0; // Gaussian distance bias
            sc[mrow * 520 + j] = val;
        }
    }
    __syncthreads();

    // --- row max / sum ---
    if (tid < 16) {
        const float* row = sc + tid * 520;
        float m = -3.4e38f;
        for (int j = 0; j < LL; ++j) m = fmaxf(m, row[j]);
        float s = 0.0f;
        for (int j = 0; j < LL; ++j) s += __expf(row[j] - m);
        rmax[tid] = m;
        rinv[tid] = 1.0f / s;
    }
    __syncthreads();

    // --- normalize in place ---
    {
        int row = tid >> 4;
        int cb = (tid & 15) << 5;
        float m = rmax[row], is = rinv[row];
        float* rp = sc + row * 520 + cb;
        for (int cci = 0; cci < 32; ++cci)
            rp[cci] = __expf(rp[cci] - m) * is;
    }
    __syncthreads();

    // --- PV: waves 0..3 each do one 16-col slab of HD=64 ---
    if (wave < 4) {
        int n0 = wave << 4;
        const float* prow = sc + ml * 520 + koff;
        const float* vcol = vb + n0 + ml;
        v8f o = {};
        for (int kk = 0; kk < LL; kk += 4) {
            v2f a = ld2(prow + kk);
            int kr = kk + koff;
            v2f bv;
            bv.x = vcol[kr * HD];
            bv.y = vcol[(kr + 1) * HD];
            o = wmma4(a, bv, o);
        }
        for (int r = 0; r < 8; ++r) {
            int mrow = r + mo;
            ctx[(size_t)(b * LL + i0 + mrow) * DD + h * HD + n0 + ml] = o[r];
        }
    }
}

// ---------------------------------------------------------------------------
// Kernel 4: out = ctx @ w_fc + b_fc.  Same TDM-staged macro-tile GEMM,
// 32(M) x 64(N) per block: grid 128 x 16.
__global__ void k_fc(const float* __restrict__ ctx, const float* __restrict__ w_fc,
                     const float* __restrict__ b_fc, float* __restrict__ out) {
    __shared__ float Bs[2][32 * 64];
    int tid = threadIdx.x;
    int wave = tid >> 5, lane = tid & 31;
    int mB = blockIdx.x >> 4, nB = blockIdx.x & 15;
    int m0 = mB << 5, n0 = nB << 6;
    int wm = wave >> 2, wn = wave & 3;
    int ml = lane & 15, koff = (lane >> 4) << 1, mo = (lane >> 4) << 3;

    unsigned lds0 = (unsigned)(size_t)&Bs[0][0];
    unsigned lds1 = (unsigned)(size_t)&Bs[1][0];

    if (wave == 0)
        tdm_load_tile(lds0, w_fc + n0, 64, 32, DD, DD, DD);

    const float* arow = ctx + (size_t)(m0 + (wm << 4) + ml) * DD + koff;
    v8f c = {};
    for (int kc = 0; kc < 32; ++kc) {
        if (wave == 0) {
            if (kc < 31) {
                tdm_load_tile((kc & 1) ? lds0 : lds1,
                              w_fc + (size_t)(kc + 1) * 32 * DD + n0,
                              64, 32, DD, DD, DD);
                __builtin_amdgcn_s_wait_tensorcnt(1);
            } else {
                __builtin_amdgcn_s_wait_tensorcnt(0);
            }
        }
        __syncthreads();
        const float* bs = &Bs[kc & 1][0] + (wn << 4) + ml;
        int kbase = kc << 5;
        for (int kk = 0; kk < 32; kk += 4) {
            v2f a = ld2(arow + kbase + kk);
            v2f bv;
            bv.x = bs[(kk + koff) << 6];
            bv.y = bs[(kk + koff + 1) << 6];
            c = wmma4(a, bv, c);
        }
        __syncthreads();
    }

    int ncol = n0 + (wn << 4);
    float bn = b_fc[ncol + ml];
    int mrow0 = m0 + (wm << 4);
    for (int r = 0; r < 8; ++r) {
        int m = mrow0 + r + mo;
        out[(size_t)m * DD + ncol + ml] = c[r] + bn;
    }
}

// ---------------------------------------------------------------------------
extern "C" void kernel_launch(void* const* d_in, const int* in_sizes, int n_in,
                              void* d_out, int out_size, void* d_ws, size_t ws_size,
                              hipStream_t stream) {
    const float* x        = (const float*)d_in[0];
    const int*   mask     = (const int*)d_in[1];
    const int*   qmask    = (const int*)d_in[2];
    const float* w_qkv    = (const float*)d_in[3];
    const float* w_qkpos  = (const float*)d_in[4];
    const float* w_fc     = (const float*)d_in[5];
    const float* b_fc     = (const float*)d_in[6];
    const float* shift    = (const float*)d_in[7];
    const float* bias     = (const float*)d_in[8];
    float* out = (float*)d_out;

    // workspace layout (floats): q | k | v | q_p | k_p | ctx  (~71 MB)
    float* ws  = (float*)d_ws;
    const size_t QKV = (size_t)BB * HH * LL * HD;   // 4,194,304
    const size_t PPS = (size_t)HH * LL * HD;        //   524,288
    float* qw  = ws;
    float* kw  = ws + QKV;
    float* vw  = ws + 2 * QKV;
    float* qpw = ws + 3 * QKV;
    float* kpw = ws + 3 * QKV + PPS;
    float* ctx = ws + 3 * QKV + 2 * PPS;

    k_qkv_proj<<<128 * 48, 256, 0, stream>>>(x, w_qkv, qw, kw, vw);
    k_pos<<<(LL * D2) / 256, 256, 0, stream>>>(w_qkpos, qpw, kpw);
    k_attn<<<BB * HH * 32, 256, 0, stream>>>(qw, kw, vw, qpw, kpw,
                                             mask, qmask, shift, bias, ctx);
    k_fc<<<128 * 16, 256, 0, stream>>>(ctx, w_fc, b_fc, out);
}